// MXIntSoftmax_70909910057499
// MI455X (gfx1250) — compile-verified
//
#include <hip/hip_runtime.h>
#include <cstdint>
#include <cstddef>

// MXInt softmax, 16384 x 1024 f32 -> f32.
// Roofline: 128 MiB traffic @ 23.3 TB/s => ~5.6 us floor (working set fits the
// 192 MB L2, so replays run out of L2). Round-1 body was ~2.3K VALU per 4KB row
// => VALU-limited; this version cuts per-element ALU via integer range
// reduction, raw v_exp_f32, and an analytic collapse of the double requantize.
// Data movement: gfx1250 async DMA (global_load_async_to_lds_b128 +
// s_wait_asynccnt) double-buffered row streaming, B128 loads/stores,
// wave32 shuffle reductions. No matmul in this op => WMMA is not applicable.

#define FEAT        1024
#define WAVES_PB    8
#define ROWS_PW     4
#define THREADS_PB  (WAVES_PB * 32)

#if __has_builtin(__builtin_amdgcn_exp2f)
#define EXP2_RAW(x) __builtin_amdgcn_exp2f(x)   // v_exp_f32, arg in [0,1)
#else
#define EXP2_RAW(x) exp2f(x)
#endif

// input quant (mxint_quant(x,8,4)) + hw_exp; returns (mexp[7b]<<8)|(n+8)[4b]
__device__ __forceinline__ uint32_t quant_exp_elem(float x, int &nmax) {
    // e = clip(ceil(log2|x|), -8, 7), exact via exponent/mantissa fields
    uint32_t u = __float_as_uint(x) & 0x7FFFFFFFu;
    int e = (int)(u >> 23) - 127 + ((u & 0x7FFFFFu) ? 1 : 0);
    e = min(max(e, -8), 7);
    // m = clip(rne(x * 2^(7-e)), -128, 127); v_ldexp + v_cvt_i32(RNE) + clamps
    int mi = __float2int_rn(ldexpf(x, 7 - e));
    mi = min(max(mi, -128), 127);
    // hw_exp range reduction (qlog2e = 92/64 = 1.4375 exactly):
    //   floor(qx*1.4375*128) = (m*184) >> (7-e)   [arith shift == floor, exact]
    int Ti = (mi * 184) >> (7 - e);
    Ti = min(max(Ti, -1024), 1023);            // integer_floor_quantizer(.,11,7)
    int ni = Ti >> 7;                          // n = floor(new_mx) in [-8,7]
    float r = (float)(Ti & 127) * 0.0078125f;  // frac(new_mx) in [0,1), exact
    int me = __float2int_rn(ldexpf(EXP2_RAW(r), 6));  // round(2^r * 64)
    me = min(me, 127);                         // clip int8 (128 -> 127)
    nmax = max(nmax, ni);
    return ((uint32_t)me << 8) | (uint32_t)(ni + 8);
}

// mout = floor(mexp*2^12 / sum) in f32 (as reference), then
// mxint_quant(mout*2^(n-emax-8), 8,4) applied twice, collapsed analytically:
// second quant is identity except m1==64 with unclipped exponent, which
// re-quantizes the power-of-two to 127*2^(e1-8).
__device__ __forceinline__ float final_elem(uint32_t pk, float fsum, int nmax) {
    int me = (int)(pk >> 8);
    int ni = (int)(pk & 15u) - 8;
    float moutf = floorf((float)(me << 12) / fsum);   // f32 div like reference
    int mout = (int)moutf;                            // exact, in [0, 508]
    int k = ni - nmax - 8;                            // in [-23, -8]
    int cl = 32 - __clz(mout - 1);                    // ceil(log2(mout)), mout>=1
    int e1 = max(cl + k, -8);                         // never exceeds +7 here
    int sh = k + 7 - e1;
    int m1 = __float2int_rn(ldexpf(moutf, sh));       // rne(mout*2^sh), exact fp
    m1 = min(m1, 127);
    bool fix = (m1 == 64) && (e1 > -8);               // power-of-two reclip case
    int m2  = fix ? 127 : m1;
    int sh2 = e1 - (fix ? 8 : 7);
    return ldexpf((float)m2, sh2);                    // mout==0 path yields 0
}

__device__ __forceinline__ void wait_async_le8() {
    asm volatile("s_wait_asynccnt 8" ::: "memory");
}
__device__ __forceinline__ void wait_async_0() {
    asm volatile("s_wait_asynccnt 0" ::: "memory");
}

__global__ __launch_bounds__(THREADS_PB)
void mxint_softmax_kernel(const float* __restrict__ x,
                          float* __restrict__ out, int nrows)
{
    // per-wave private double buffer: 2 x 4KB rows; 64KB/block, no barriers
    __shared__ __align__(16) float sh[WAVES_PB][2][FEAT];

    const int lane = (int)(threadIdx.x & 31u);
    const int wv   = (int)(threadIdx.x >> 5);
    const int row0 = ((int)blockIdx.x * WAVES_PB + wv) * ROWS_PW;

    // low 32 bits of a flat shared pointer == LDS byte offset (LDS aperture)
    const uint32_t lds0 = (uint32_t)(uintptr_t)(&sh[wv][0][0]) + (uint32_t)(lane * 16);

    // async DMA one 4KB row into buffer `buf`: 8 x B128 per lane (512B/instr)
    auto prefetch = [&](int buf, int row) {
        uint64_t g = (uint64_t)(uintptr_t)(x + (size_t)row * FEAT) + (uint64_t)(lane * 16);
        uint32_t l = lds0 + (uint32_t)(buf * (FEAT * 4));
        asm volatile(
            "global_load_async_to_lds_b128 %0, %1, off\n\t"
            "global_load_async_to_lds_b128 %0, %1, off offset:512\n\t"
            "global_load_async_to_lds_b128 %0, %1, off offset:1024\n\t"
            "global_load_async_to_lds_b128 %0, %1, off offset:1536\n\t"
            "global_load_async_to_lds_b128 %0, %1, off offset:2048\n\t"
            "global_load_async_to_lds_b128 %0, %1, off offset:2560\n\t"
            "global_load_async_to_lds_b128 %0, %1, off offset:3072\n\t"
            "global_load_async_to_lds_b128 %0, %1, off offset:3584"
            :: "v"(l), "v"(g) : "memory");
    };

    if (row0 >= nrows) return;
    prefetch(0, row0);

    for (int r = 0; r < ROWS_PW; ++r) {
        const int row = row0 + r;
        if (row >= nrows) break;

        const bool pf = (r + 1 < ROWS_PW) && (row + 1 < nrows);
        if (pf) prefetch((r + 1) & 1, row + 1);        // overlap DMA w/ compute
        if (pf) wait_async_le8(); else wait_async_0(); // in-order retire

        const float* sb = &sh[wv][r & 1][0];

        // ---- pass A: quantize + hw_exp; packed (mexp, n) stays in VGPRs ----
        uint32_t pk[32];
        int nmax = -8;
        #pragma unroll
        for (int j = 0; j < 8; ++j) {
            const float4 v = *(const float4*)(sb + j * 128 + lane * 4);  // ds_load_b128
            pk[j * 4 + 0] = quant_exp_elem(v.x, nmax);
            pk[j * 4 + 1] = quant_exp_elem(v.y, nmax);
            pk[j * 4 + 2] = quant_exp_elem(v.z, nmax);
            pk[j * 4 + 3] = quant_exp_elem(v.w, nmax);
        }

        // ---- row max exponent (wave32 shuffle reduction) ----
        #pragma unroll
        for (int off = 16; off > 0; off >>= 1) {
            int o = __shfl_xor(nmax, off, 32);
            nmax = o > nmax ? o : nmax;
        }

        // ---- BFP mantissa sum: sum_i floor(mexp_i*16 >> (emax - n_i)) ----
        const int base = nmax + 8;
        int ssum = 0;
        #pragma unroll
        for (int j = 0; j < 32; ++j) {
            int me4 = (int)((pk[j] >> 4) & 0x7F0u);      // mexp << 4
            int shf = base - (int)(pk[j] & 15u);         // (emax - n) in [0,15]
            ssum += me4 >> shf;
        }
        #pragma unroll
        for (int off = 16; off > 0; off >>= 1)
            ssum += __shfl_xor(ssum, off, 32);
        const float fsum = (float)ssum;                  // <= 2^21, exact

        // ---- pass B: divide, scale, collapsed double-quant, B128 store ----
        float* ob = out + (size_t)row * FEAT;
        #pragma unroll
        for (int j = 0; j < 8; ++j) {
            float4 o;
            o.x = final_elem(pk[j * 4 + 0], fsum, nmax);
            o.y = final_elem(pk[j * 4 + 1], fsum, nmax);
            o.z = final_elem(pk[j * 4 + 2], fsum, nmax);
            o.w = final_elem(pk[j * 4 + 3], fsum, nmax);
            *(float4*)(ob + j * 128 + lane * 4) = o;     // global_store_b128
        }
    }
}

extern "C" void kernel_launch(void* const* d_in, const int* in_sizes, int n_in,
                              void* d_out, int out_size, void* d_ws, size_t ws_size,
                              hipStream_t stream) {
    (void)n_in; (void)d_ws; (void)ws_size; (void)out_size;
    const float* x = (const float*)d_in[0];
    float* out = (float*)d_out;
    const int nrows = in_sizes[0] / FEAT;                     // 16384
    const int rows_per_block = WAVES_PB * ROWS_PW;            // 32
    const int grid = (nrows + rows_per_block - 1) / rows_per_block;  // 512
    mxint_softmax_kernel<<<grid, THREADS_PB, 0, stream>>>(x, out, nrows);
}